// Net_MP_11390253269729
// MI455X (gfx1250) — compile-verified
//
#include <hip/hip_runtime.h>
#include <hip/hip_bf16.h>

typedef __attribute__((ext_vector_type(2))) float v2f;
typedef __attribute__((ext_vector_type(8))) float v8f;

#define WMMA_F32X4(A, B, C) \
    __builtin_amdgcn_wmma_f32_16x16x4_f32(false, (A), false, (B), (short)0, (C), false, false)

// ---------------------------------------------------------------------------
// zero fill
// ---------------------------------------------------------------------------
__global__ void zero_f32(float* __restrict__ p, int n) {
    int t = blockIdx.x * blockDim.x + threadIdx.x;
    if (t < n) p[t] = 0.0f;
}

// ---------------------------------------------------------------------------
// Layer 1 edge messages (in=2): scalar, one thread per (edge, out-channel).
// msg[e,o] = sum_i x[src,i] * (ea0*nnW[(i*32+o),0] + ea1*nnW[(i*32+o),1] + nnb[i*32+o])
// ---------------------------------------------------------------------------
__global__ void edge_msg_l1(const float* __restrict__ x,
                            const int* __restrict__ src,
                            const int* __restrict__ dst,
                            const float* __restrict__ ea,
                            const float* __restrict__ nnW,   // [64,2]
                            const float* __restrict__ nnb,   // [64]
                            float* __restrict__ aggr,        // [N,32]
                            int E) {
    int t = blockIdx.x * blockDim.x + threadIdx.x;
    if (t >= E * 32) return;
    int e = t >> 5;
    int o = t & 31;
    float ea0 = ea[e * 2 + 0];
    float ea1 = ea[e * 2 + 1];
    int s = src[e];
    float x0 = x[s * 2 + 0];
    float x1 = x[s * 2 + 1];
    int r0 = o;       // i=0 row
    int r1 = 32 + o;  // i=1 row
    float w0 = fmaf(ea0, nnW[r0 * 2 + 0], fmaf(ea1, nnW[r0 * 2 + 1], nnb[r0]));
    float w1 = fmaf(ea0, nnW[r1 * 2 + 0], fmaf(ea1, nnW[r1 * 2 + 1], nnb[r1]));
    atomicAdd(&aggr[dst[e] * 32 + o], fmaf(x0, w0, x1 * w1));
}

// ---------------------------------------------------------------------------
// Layer 1 node update: h = relu(aggr + x @ root1 + b1), root1 is [2,32]
// ---------------------------------------------------------------------------
__global__ void node_update_l1(const float* __restrict__ x,
                               const float* __restrict__ aggr,
                               const float* __restrict__ root,  // [2,32]
                               const float* __restrict__ bias,  // [32]
                               float* __restrict__ hout,
                               int N) {
    int t = blockIdx.x * blockDim.x + threadIdx.x;
    if (t >= N * 32) return;
    int i = t >> 5;
    int o = t & 31;
    float y = aggr[t] + x[i * 2 + 0] * root[o] + x[i * 2 + 1] * root[32 + o] + bias[o];
    hout[t] = fmaxf(y, 0.0f);
}

// ---------------------------------------------------------------------------
// Layers 2/3 edge messages via V_WMMA_F32_16X16X4_F32.
// One wave handles a 16-edge tile.  For the tile:
//   Y0 = Xj @ A0, Y1 = Xj @ A1, Y2 = Xj @ B   (A0/A1/B are [32,32] derived from nn_W/nn_b)
//   msg = ea0 * Y0 + ea1 * Y1 + Y2;  atomic scatter-add into aggr[dst].
// A-fragment layout (16x4 f32): lane%16 = M row; lanes<16 hold K,K+1 = kc*4+{0,1},
// lanes>=16 hold kc*4+{2,3}.  B mirrors with lane%16 = N col.
// C/D layout: VGPR v = row v (lanes 0-15) / v+8 (lanes 16-31), col = lane%16 (+16 for hi half).
// ---------------------------------------------------------------------------
__global__ __launch_bounds__(128) void edge_msg_wmma(
        const float* __restrict__ h,    // [N,32]
        const int* __restrict__ src,    // [E]
        const int* __restrict__ dst,    // [E]
        const float* __restrict__ ea,   // [E,2]
        const float* __restrict__ nnW,  // [1024,2]
        const float* __restrict__ nnb,  // [1024]
        float* __restrict__ aggr,       // [N,32]
        int E) {
    const int lane = threadIdx.x & 31;
    const int wave = threadIdx.x >> 5;
    const int tile = blockIdx.x * 4 + wave;
    const int e0 = tile * 16;
    if (e0 >= E) return;  // wave-uniform: EXEC stays all-ones for live waves

    const int m = lane & 15;             // A row / B col within tile
    const int khalf = (lane >> 4) * 2;   // 0 or 2: which K pair this lane holds

    // Gather source-node row for this lane's A-matrix row (clamped for safety).
    int eg = e0 + m;
    if (eg >= E) eg = E - 1;
    const float* hrow = h + (long)src[eg] * 32;

    v8f acc0a = {}, acc0b = {};  // Y0, lo/hi output halves
    v8f acc1a = {}, acc1b = {};  // Y1
    v8f acc2a = {}, acc2b = {};  // Y2

#pragma unroll
    for (int kc = 0; kc < 8; ++kc) {
        const int kb = kc * 4 + khalf;
        v2f a;
        a.x = hrow[kb + 0];
        a.y = hrow[kb + 1];
#pragma unroll
        for (int nh = 0; nh < 2; ++nh) {
            const int n = nh * 16 + m;
            const int r0 = (kb + 0) * 32 + n;  // nn_W row index = i*32 + o
            const int r1 = (kb + 1) * 32 + n;
            v2f b0, b1, b2;
            b0.x = nnW[r0 * 2 + 0]; b0.y = nnW[r1 * 2 + 0];  // A0 frag
            b1.x = nnW[r0 * 2 + 1]; b1.y = nnW[r1 * 2 + 1];  // A1 frag
            b2.x = nnb[r0];         b2.y = nnb[r1];          // B  frag
            if (nh == 0) {
                acc0a = WMMA_F32X4(a, b0, acc0a);
                acc1a = WMMA_F32X4(a, b1, acc1a);
                acc2a = WMMA_F32X4(a, b2, acc2a);
            } else {
                acc0b = WMMA_F32X4(a, b0, acc0b);
                acc1b = WMMA_F32X4(a, b1, acc1b);
                acc2b = WMMA_F32X4(a, b2, acc2b);
            }
        }
    }

    // Combine with per-edge scalars and scatter-add.
    const int rbase = (lane >> 4) * 8;
#pragma unroll
    for (int v = 0; v < 8; ++v) {
        const int row = rbase + v;
        const int e = e0 + row;
        if (e < E) {
            const float s0 = ea[e * 2 + 0];
            const float s1 = ea[e * 2 + 1];
            const int d = dst[e];
            const float mA = fmaf(s0, acc0a[v], fmaf(s1, acc1a[v], acc2a[v]));
            const float mB = fmaf(s0, acc0b[v], fmaf(s1, acc1b[v], acc2b[v]));
            atomicAdd(&aggr[d * 32 + m], mA);
            atomicAdd(&aggr[d * 32 + 16 + m], mB);
        }
    }
}

// ---------------------------------------------------------------------------
// Layers 2/3 node update via WMMA: hout = relu(aggr + hprev @ root + bias).
// aggr is loaded directly into the C-operand layout; root GEMM runs on WMMA.
// ---------------------------------------------------------------------------
__global__ __launch_bounds__(128) void node_update_wmma(
        const float* __restrict__ hprev,  // [N,32]
        const float* __restrict__ aggr,   // [N,32]
        const float* __restrict__ root,   // [32,32] (k,o)
        const float* __restrict__ bias,   // [32]
        float* __restrict__ hout,         // [N,32]
        int N) {
    const int lane = threadIdx.x & 31;
    const int wave = threadIdx.x >> 5;
    const int tile = blockIdx.x * 4 + wave;
    const int n0 = tile * 16;
    if (n0 >= N) return;

    const int m = lane & 15;
    const int khalf = (lane >> 4) * 2;
    const int rbase = (lane >> 4) * 8;

    v8f accA, accB;
#pragma unroll
    for (int v = 0; v < 8; ++v) {
        int row = n0 + rbase + v;
        if (row >= N) row = N - 1;
        accA[v] = aggr[row * 32 + m];
        accB[v] = aggr[row * 32 + 16 + m];
    }

    int ag = n0 + m;
    if (ag >= N) ag = N - 1;
    const float* hrow = hprev + (long)ag * 32;

#pragma unroll
    for (int kc = 0; kc < 8; ++kc) {
        const int kb = kc * 4 + khalf;
        v2f a;
        a.x = hrow[kb + 0];
        a.y = hrow[kb + 1];
        v2f b0, b1;
        b0.x = root[(kb + 0) * 32 + m];      b0.y = root[(kb + 1) * 32 + m];
        b1.x = root[(kb + 0) * 32 + 16 + m]; b1.y = root[(kb + 1) * 32 + 16 + m];
        accA = WMMA_F32X4(a, b0, accA);
        accB = WMMA_F32X4(a, b1, accB);
    }

#pragma unroll
    for (int v = 0; v < 8; ++v) {
        const int row = n0 + rbase + v;
        if (row < N) {
            hout[row * 32 + m]      = fmaxf(accA[v] + bias[m], 0.0f);
            hout[row * 32 + 16 + m] = fmaxf(accB[v] + bias[16 + m], 0.0f);
        }
    }
}

// ---------------------------------------------------------------------------
// Dense head: out = fc2(relu(fc1(h)))   (trivial FLOPs; scalar)
// ---------------------------------------------------------------------------
__global__ void head_kernel(const float* __restrict__ h,     // [N,32]
                            const float* __restrict__ fc1W,  // [32,32] (o,k)
                            const float* __restrict__ fc1b,  // [32]
                            const float* __restrict__ fc2W,  // [1,32]
                            const float* __restrict__ fc2b,  // [1]
                            float* __restrict__ out, int N) {
    int i = blockIdx.x * blockDim.x + threadIdx.x;
    if (i >= N) return;
    float hr[32];
#pragma unroll
    for (int k = 0; k < 32; ++k) hr[k] = h[i * 32 + k];
    float o = fc2b[0];
#pragma unroll 4
    for (int oo = 0; oo < 32; ++oo) {
        float y = fc1b[oo];
#pragma unroll
        for (int k = 0; k < 32; ++k) y = fmaf(hr[k], fc1W[oo * 32 + k], y);
        o = fmaf(fc2W[oo], fmaxf(y, 0.0f), o);
    }
    out[i] = o;
}

// ---------------------------------------------------------------------------
// launch
// ---------------------------------------------------------------------------
extern "C" void kernel_launch(void* const* d_in, const int* in_sizes, int n_in,
                              void* d_out, int out_size, void* d_ws, size_t ws_size,
                              hipStream_t stream) {
    const float* x     = (const float*)d_in[0];
    const int*   eidx  = (const int*)  d_in[1];
    const float* ea    = (const float*)d_in[2];
    const float* nn1W  = (const float*)d_in[3];
    const float* nn1b  = (const float*)d_in[4];
    const float* root1 = (const float*)d_in[5];
    const float* b1    = (const float*)d_in[6];
    const float* nn2W  = (const float*)d_in[7];
    const float* nn2b  = (const float*)d_in[8];
    const float* root2 = (const float*)d_in[9];
    const float* b2    = (const float*)d_in[10];
    const float* nn3W  = (const float*)d_in[11];
    const float* nn3b  = (const float*)d_in[12];
    const float* root3 = (const float*)d_in[13];
    const float* b3    = (const float*)d_in[14];
    const float* fc1W  = (const float*)d_in[15];
    const float* fc1b  = (const float*)d_in[16];
    const float* fc2W  = (const float*)d_in[17];
    const float* fc2b  = (const float*)d_in[18];

    const int N = in_sizes[0] / 2;   // x is [N,2]
    const int E = in_sizes[2] / 2;   // edge_attr is [E,2]
    const int* src = eidx;
    const int* dst = eidx + E;

    float* aggr = (float*)d_ws;        // [N,32]
    float* bufA = aggr + (size_t)N * 32;
    float* bufB = bufA + (size_t)N * 32;

    const int NC = N * 32;
    const int zgrid = (NC + 255) / 256;
    const int etiles = (E + 15) / 16;
    const int eblocks = (etiles + 3) / 4;   // 4 waves / block
    const int ntiles = (N + 15) / 16;
    const int nblocks = (ntiles + 3) / 4;

    // ---- layer 1 (scalar, in=2) ----
    zero_f32<<<zgrid, 256, 0, stream>>>(aggr, NC);
    edge_msg_l1<<<(E * 32 + 255) / 256, 256, 0, stream>>>(x, src, dst, ea, nn1W, nn1b, aggr, E);
    node_update_l1<<<zgrid, 256, 0, stream>>>(x, aggr, root1, b1, bufA, N);

    // ---- layer 2 (WMMA) ----
    zero_f32<<<zgrid, 256, 0, stream>>>(aggr, NC);
    edge_msg_wmma<<<eblocks, 128, 0, stream>>>(bufA, src, dst, ea, nn2W, nn2b, aggr, E);
    node_update_wmma<<<nblocks, 128, 0, stream>>>(bufA, aggr, root2, b2, bufB, N);

    // ---- layer 3 (WMMA) ----
    zero_f32<<<zgrid, 256, 0, stream>>>(aggr, NC);
    edge_msg_wmma<<<eblocks, 128, 0, stream>>>(bufB, src, dst, ea, nn3W, nn3b, aggr, E);
    node_update_wmma<<<nblocks, 128, 0, stream>>>(bufB, aggr, root3, b3, bufA, N);

    // ---- head ----
    head_kernel<<<(N + 255) / 256, 256, 0, stream>>>(bufA, fc1W, fc1b, fc2W, fc2b,
                                                     (float*)d_out, N);
}